// SoftEmbeddedDecisionRules_56023553409032
// MI455X (gfx1250) — compile-verified
//
#include <hip/hip_runtime.h>

// Hierarchical 2-way-softmax tree over 1024 classes, one row per wave32.
// Memory-bound: 256 MB traffic -> ~11 us floor at 23.3 TB/s on MI455X.
// Uses CDNA5 async global->LDS data movement (ASYNCcnt) for coalesced staging.

#define ROW_C 1024
#define WAVES_PER_BLOCK 8

__global__ __launch_bounds__(256) void
soft_tree_kernel(const float* __restrict__ in, float* __restrict__ out, int rows) {
    __shared__ __align__(16) float smem[WAVES_PER_BLOCK * ROW_C];

    const int wave = threadIdx.x >> 5;
    const int lane = threadIdx.x & 31;
    const int row  = blockIdx.x * WAVES_PER_BLOCK + wave;
    if (row >= rows) return;

    const float* rowIn  = in  + (size_t)row * ROW_C;
    float*       rowOut = out + (size_t)row * ROW_C;
    float*       lrow   = smem + wave * ROW_C;

    // ---- async coalesced copy: global row (4KB) -> LDS, identity layout ----
    // Generic LDS pointer: low 32 bits are the workgroup-relative LDS offset.
    unsigned    ldsBase = (unsigned)(size_t)lrow + (unsigned)(lane * 16);
    const char* gBase   = (const char*)rowIn + lane * 16;
#pragma unroll
    for (int i = 0; i < 8; ++i) {
        unsigned    laddr = ldsBase + (unsigned)(i * 512);
        const char* gaddr = gBase + i * 512;
        asm volatile("global_load_async_to_lds_b128 %0, %1, off"
                     :: "v"(laddr), "v"(gaddr) : "memory");
    }
    asm volatile("s_wait_asynccnt 0" ::: "memory");

    // ---- per-lane contiguous 32 classes from LDS ----
    float v[32];
    const float4* lptr = (const float4*)(lrow + lane * 32);
#pragma unroll
    for (int j = 0; j < 8; ++j) {
        float4 q = lptr[j];
        v[4 * j + 0] = q.x; v[4 * j + 1] = q.y;
        v[4 * j + 2] = q.z; v[4 * j + 3] = q.w;
    }

    // ---- intra-lane block sums (tree levels seg = 2,4,8,16,32) ----
    float s2[16], s4[8], s8[4], s16[2], s32v;
#pragma unroll
    for (int k = 0; k < 16; ++k) s2[k] = v[2 * k] + v[2 * k + 1];
#pragma unroll
    for (int k = 0; k < 8; ++k)  s4[k] = s2[2 * k] + s2[2 * k + 1];
#pragma unroll
    for (int k = 0; k < 4; ++k)  s8[k] = s4[2 * k] + s4[2 * k + 1];
    s16[0] = s8[0] + s8[1];
    s16[1] = s8[2] + s8[3];
    s32v   = s16[0] + s16[1];

    // 2-way softmax pair: e = exp(hi-lo), f_lo = 1/(1+e), f_hi = 1 - f_lo
    float p[32];
    // seg = 1
#pragma unroll
    for (int k = 0; k < 16; ++k) {
        float e   = __expf(v[2 * k + 1] - v[2 * k]);
        float inv = __builtin_amdgcn_rcpf(1.0f + e);
        p[2 * k]     = inv;
        p[2 * k + 1] = 1.0f - inv;
    }
    // seg = 2
#pragma unroll
    for (int k = 0; k < 8; ++k) {
        float e   = __expf((s2[2 * k + 1] - s2[2 * k]) * 0.5f);
        float inv = __builtin_amdgcn_rcpf(1.0f + e);
        float b   = 1.0f - inv;
        p[4 * k + 0] *= inv; p[4 * k + 1] *= inv;
        p[4 * k + 2] *= b;   p[4 * k + 3] *= b;
    }
    // seg = 4
#pragma unroll
    for (int k = 0; k < 4; ++k) {
        float e   = __expf((s4[2 * k + 1] - s4[2 * k]) * 0.25f);
        float inv = __builtin_amdgcn_rcpf(1.0f + e);
        float b   = 1.0f - inv;
#pragma unroll
        for (int t = 0; t < 4; ++t) { p[8 * k + t] *= inv; p[8 * k + 4 + t] *= b; }
    }
    // seg = 8
#pragma unroll
    for (int k = 0; k < 2; ++k) {
        float e   = __expf((s8[2 * k + 1] - s8[2 * k]) * 0.125f);
        float inv = __builtin_amdgcn_rcpf(1.0f + e);
        float b   = 1.0f - inv;
#pragma unroll
        for (int t = 0; t < 8; ++t) { p[16 * k + t] *= inv; p[16 * k + 8 + t] *= b; }
    }
    // seg = 16
    {
        float e   = __expf((s16[1] - s16[0]) * 0.0625f);
        float inv = __builtin_amdgcn_rcpf(1.0f + e);
        float b   = 1.0f - inv;
#pragma unroll
        for (int t = 0; t < 16; ++t) { p[t] *= inv; p[16 + t] *= b; }
    }

    // ---- cross-lane levels (seg = 32,64,128,256,512) via shfl_xor butterfly ----
    float coarse = 1.0f;
    float s  = s32v;
    float rs = 1.0f / 32.0f;
#pragma unroll
    for (int m = 1; m <= 16; m <<= 1) {
        float sib = __shfl_xor(s, m, 32);
        float e   = __expf((sib - s) * rs);       // sigmoid(mine - sib)
        coarse   *= __builtin_amdgcn_rcpf(1.0f + e);
        s  += sib;                                // group sum at next scale
        rs *= 0.5f;
    }

#pragma unroll
    for (int j = 0; j < 32; ++j) p[j] *= coarse;

    // ---- store 32 contiguous class probs per lane ----
    float4* optr = (float4*)(rowOut + lane * 32);
#pragma unroll
    for (int j = 0; j < 8; ++j) {
        float4 q;
        q.x = p[4 * j + 0]; q.y = p[4 * j + 1];
        q.z = p[4 * j + 2]; q.w = p[4 * j + 3];
        optr[j] = q;
    }
}

extern "C" void kernel_launch(void* const* d_in, const int* in_sizes, int n_in,
                              void* d_out, int out_size, void* d_ws, size_t ws_size,
                              hipStream_t stream) {
    const float* in  = (const float*)d_in[0];
    float*       out = (float*)d_out;
    const int rows   = in_sizes[0] / ROW_C;            // 32768
    const int blocks = (rows + WAVES_PER_BLOCK - 1) / WAVES_PER_BLOCK;
    soft_tree_kernel<<<blocks, 256, 0, stream>>>(in, out, rows);
}